// CBOWModel_24687472017957
// MI455X (gfx1250) — compile-verified
//
#include <hip/hip_runtime.h>

// CBOW negative-sampling loss for MI455X (gfx1250, wave32).
// Memory-bound gather (~370 MB reads -> ~16us floor @ 23.3 TB/s).
//   - w-embedding gathers: TDM TENSOR_LOAD_TO_LDS in gather mode (8 rows /
//     descriptor, 32-bit row indices, LDS pad 256+8 DWORDs = 264-float stride).
//   - u-embedding gathers: coalesced float4 vector loads + ctx-sum (needs
//     accumulation, so VGPR path; overlaps the in-flight TDM DMAs).
//   - per-row dot products: V_WMMA_F32_16X16X4_F32 with ones B-matrix
//     (every column of D = row sum of A = u ⊙ w), 64 accumulating WMMAs.

#define EMB    256
#define CTX    10
#define TILE   16
#define STRIDE 264   // floats per LDS row: 16B-aligned; matches TDM pad 256+8

typedef __attribute__((ext_vector_type(2))) float        v2f;
typedef __attribute__((ext_vector_type(8))) float        v8f;
typedef __attribute__((ext_vector_type(4))) unsigned int v4u;
typedef __attribute__((ext_vector_type(4))) int          v4i;
typedef __attribute__((ext_vector_type(8))) int          v8i;

__device__ __forceinline__ float log_sigmoid(float x) {
    // stable: min(x,0) - log1p(exp(-|x|))
    return fminf(x, 0.0f) - log1pf(expf(-fabsf(x)));
}

__global__ void cbow_zero_out(float* out) { out[0] = 0.0f; }

__global__ __launch_bounds__(256)
void cbow_loss_kernel(const int* __restrict__ pos_u, const int* __restrict__ pos_w,
                      const int* __restrict__ neg_u, const int* __restrict__ neg_w,
                      const float* __restrict__ u_weight, const float* __restrict__ w_weight,
                      float* __restrict__ out)
{
    __shared__ float sU[2][TILE][STRIDE];   // [pos/neg] ctx-summed u embeddings
    __shared__ float sW[2][TILE][STRIDE];   // [pos/neg] w embeddings (TDM-filled)
    __shared__ int   sIdxU[2][TILE][CTX];

    const int tid  = threadIdx.x;
    const int wave = tid >> 5;
    const int lane = tid & 31;
    const int row0 = blockIdx.x * TILE;

    // ---- Issue TDM gather-mode loads for the w rows (wave 0 only: TDM
    //      ignores EXEC, so guard at wave granularity to avoid duplicates). ----
    if (wave == 0) {
        const unsigned long long gaddr = (unsigned long long)(uintptr_t)w_weight;
        #pragma unroll
        for (int pn = 0; pn < 2; ++pn) {
            const int* widx = (pn == 0 ? pos_w : neg_w) + row0;
            #pragma unroll
            for (int half = 0; half < 2; ++half) {
                const unsigned int lds_off =
                    (unsigned int)(uintptr_t)&sW[pn][half * 8][0];
                // D# group 0: count=1, gather_mode=1, 32-bit indices,
                //             lds_addr, 57-bit global base, type=2 (image)
                v4u g0;
                g0[0] = 1u | (1u << 30) | (1u << 31);
                g0[1] = lds_off;
                g0[2] = (unsigned int)(gaddr & 0xFFFFFFFFu);
                g0[3] = (unsigned int)((gaddr >> 32) & 0x01FFFFFFu) | (2u << 30);
                // D# group 1: data_size=4B (2), pad_enable, pad_interval=7
                //             (256 DW), pad_amount=7 (8 DW) -> 264-float stride;
                //             tensor_dim0=256, tensor_dim1=199999, tile_dim0=256,
                //             tile_dim1=8 valid indices, dim0_stride=256.
                v8i g1;
                g1[0] = (2 << 16) | (1 << 20) | (7 << 22) | (7 << 25);
                g1[1] = (int)(256u << 16);                       // dim0 lo16
                g1[2] = (int)((199999u & 0xFFFFu) << 16);        // dim0 hi=0 | dim1 lo16
                g1[3] = (int)((256u << 16) | (199999u >> 16));   // tile_dim0 | dim1 hi16
                g1[4] = 8;                                       // tile_dim1 = #indices
                g1[5] = 256;                                     // dim0_stride lo32
                g1[6] = 0;
                g1[7] = 0;
                // D# groups 2/3: eight 32-bit row indices
                v4i g2, g3;
                g2[0] = widx[half * 8 + 0]; g2[1] = widx[half * 8 + 1];
                g2[2] = widx[half * 8 + 2]; g2[3] = widx[half * 8 + 3];
                g3[0] = widx[half * 8 + 4]; g3[1] = widx[half * 8 + 5];
                g3[2] = widx[half * 8 + 6]; g3[3] = widx[half * 8 + 7];
                // 6-arg toolchain variant: extra 256-bit group, zero-filled.
                v8i g4 = {0, 0, 0, 0, 0, 0, 0, 0};
                __builtin_amdgcn_tensor_load_to_lds(g0, g1, g2, g3, g4,
                                                    /*cpol=*/0);
            }
        }
    }

    // Preload u indices into LDS.
    if (tid < TILE * CTX) {
        sIdxU[0][tid / CTX][tid % CTX] = pos_u[row0 * CTX + tid];
        sIdxU[1][tid / CTX][tid % CTX] = neg_u[row0 * CTX + tid];
    }
    __syncthreads();

    // ---- Phase 1: coalesced u gather + context sum (overlaps TDM DMAs) ----
    for (int j = tid; j < TILE * (EMB / 4); j += 256) {
        const int row = j / (EMB / 4);
        const int q   = j % (EMB / 4);
        #pragma unroll
        for (int pn = 0; pn < 2; ++pn) {
            float4 acc = make_float4(0.f, 0.f, 0.f, 0.f);
            #pragma unroll
            for (int c = 0; c < CTX; ++c) {
                const float4 v =
                    *(const float4*)(u_weight + (size_t)sIdxU[pn][row][c] * EMB + q * 4);
                acc.x += v.x; acc.y += v.y; acc.z += v.z; acc.w += v.w;
            }
            *(float4*)(&sU[pn][row][q * 4]) = acc;
        }
    }

    // Drain wave 0's TDM ops, then publish to the workgroup.
    if (wave == 0) __builtin_amdgcn_s_wait_tensorcnt(0);
    __syncthreads();

    // ---- Phase 2: row-dot via WMMA f32 16x16x4, B = ones ----
    // A layout (32-bit A 16x4): lanes 0-15 hold M=lane, K = 4s+{0,1};
    //                           lanes 16-31 hold M=lane-16, K = 4s+{2,3}.
    // D layout: c[r] = D[M=r][N=lane] (lanes 0-15), D[M=8+r][N=lane-16] (16-31).
    if (wave < 2) {                       // wave-uniform branch: EXEC stays all-1s
        const float* u = &sU[wave][0][0];
        const float* w = &sW[wave][0][0];
        const int row  = lane & 15;
        const int koff = (lane < 16) ? 0 : 2;

        v2f b; b[0] = 1.0f; b[1] = 1.0f;  // ones matrix: D[m,n] = rowsum(A[m,:])
        v8f c = {};
        #pragma unroll 4
        for (int s = 0; s < EMB / 4; ++s) {
            const int col = s * 4 + koff;
            v2f a;
            a[0] = u[row * STRIDE + col]     * w[row * STRIDE + col];
            a[1] = u[row * STRIDE + col + 1] * w[row * STRIDE + col + 1];
            c = __builtin_amdgcn_wmma_f32_16x16x4_f32(
                    /*neg_a=*/false, a, /*neg_b=*/false, b,
                    /*c_mod=*/(short)0, c, /*reuse_a=*/false, /*reuse_b=*/false);
        }

        const float sgn = (wave == 0) ? 1.0f : -1.0f;  // pos: +score, neg: -score
        float partial = 0.0f;
        #pragma unroll
        for (int r = 0; r < 8; ++r) partial += log_sigmoid(sgn * c[r]);
        // lanes 0-15 hold rows 0-7 sum, lanes 16-31 hold rows 8-15 sum; combine.
        partial += __shfl_xor(partial, 16, 32);
        if (lane == 0) atomicAdd(out, -partial);       // loss = -(sum of scores)
    }
}

extern "C" void kernel_launch(void* const* d_in, const int* in_sizes, int n_in,
                              void* d_out, int out_size, void* d_ws, size_t ws_size,
                              hipStream_t stream) {
    const int*   pos_u    = (const int*)d_in[0];
    const int*   pos_w    = (const int*)d_in[1];
    const int*   neg_u    = (const int*)d_in[2];
    const int*   neg_w    = (const int*)d_in[3];
    const float* u_weight = (const float*)d_in[4];
    const float* w_weight = (const float*)d_in[5];
    float*       out      = (float*)d_out;

    const int B = in_sizes[1];            // pos_w has B elements

    cbow_zero_out<<<1, 1, 0, stream>>>(out);
    cbow_loss_kernel<<<B / TILE, 256, 0, stream>>>(
        pos_u, pos_w, neg_u, neg_w, u_weight, w_weight, out);
}